// GaborLayer_75161927680635
// MI455X (gfx1250) — compile-verified
//
#include <hip/hip_runtime.h>
#include <math.h>

// CDNA5 / gfx1250, wave32.
typedef __attribute__((ext_vector_type(2))) float v2f;
typedef __attribute__((ext_vector_type(8))) float v8f;

#define NG 512
#define HW (512 * 512)
#define TWO_PI   6.28318530717958647692f
#define PI_F     3.14159265358979323846f
#define LOG2E    1.44269504088896340736f
#define CLIP80L2 115.415603271f   /* 80 * log2(e) */

// One block = 256 threads = 8 waves. Each wave produces 32 pixels x 3 channels.
// Per K-chunk (2 gabors): each lane evaluates its gabor at 2 pixels, feeding two
// V_WMMA_F32_16X16X4_F32 accumulations (channels = A rows, pixels = B columns).
__global__ __launch_bounds__(256) void gabor_wmma_kernel(
    const float* __restrict__ grid_x, const float* __restrict__ grid_y,
    const float* __restrict__ u, const float* __restrict__ v,
    const float* __restrict__ theta, const float* __restrict__ rel_sigma,
    const float* __restrict__ rel_freq, const float* __restrict__ gamma_,
    const float* __restrict__ psi, const float* __restrict__ amplitude,
    float* __restrict__ out)
{
    // Per-gabor geometry: uc, vc, ct, st, i2s*log2e, i2g*log2e, freq/2pi, pad
    __shared__ float sp[NG * 8];
    // Per-gabor A coefficients, interleaved per channel: cc0,cs0, cc1,cs1, cc2,cs2, 0,0
    __shared__ float sc[NG * 8];

    // ---- Parameter stage (uniform control flow; 2 gabors per thread) ----
    for (int g = threadIdx.x; g < NG; g += 256) {
        float uc = fminf(fmaxf(u[g], -1.f), 1.f);
        float vc = fminf(fmaxf(v[g], -1.f), 1.f);
        float th = fminf(fmaxf(theta[g], 0.f), 1.f) * TWO_PI;
        float ct = cosf(th), st = sinf(th);
        float rs  = fminf(fmaxf(rel_sigma[g], -5.f), 5.f);
        float sig = 0.5f + 0.5f * tanhf(rs);
        float ga  = fminf(fmaxf(gamma_[g], -5.f), 5.f);
        float gm  = 0.001f + 0.5f / (1.f + expf(-ga));

        sp[g * 8 + 0] = uc;
        sp[g * 8 + 1] = vc;
        sp[g * 8 + 2] = ct;
        sp[g * 8 + 3] = st;
        sp[g * 8 + 4] = LOG2E / (2.f * sig * sig);   // pre-scaled for v_exp (2^x)
        sp[g * 8 + 5] = LOG2E / (2.f * gm * gm);
        sp[g * 8 + 6] = expf(-rel_freq[g]);          // freq / (2*pi), for v_cos/v_sin
        sp[g * 8 + 7] = 0.f;

        // cos(f*xr + phi) = cos(phi)*cos(f*xr) - sin(phi)*sin(f*xr);  phi = psi*2*pi*pi
        for (int c = 0; c < 3; ++c) {
            float amp = 0.2f * tanhf(fminf(fmaxf(amplitude[g * 3 + c], -5.f), 5.f));
            float ph  = psi[g * 3 + c] * (TWO_PI * PI_F);
            sc[g * 8 + 2 * c]     =  amp * cosf(ph);
            sc[g * 8 + 2 * c + 1] = -amp * sinf(ph);
        }
        sc[g * 8 + 6] = 0.f;   // zero pair for padded A-rows (channels 3..15)
        sc[g * 8 + 7] = 0.f;
    }
    __syncthreads();

    // ---- Main stage ----
    const int lane   = threadIdx.x & 31;   // wave32
    const int waveid = threadIdx.x >> 5;
    const int m      = lane & 15;          // A-matrix row (channel) == pixel-in-tile
    const int hi     = lane >> 4;          // 0: K={0,1} (even gabor), 1: K={2,3} (odd)
    const int pixb   = (blockIdx.x * 8 + waveid) * 32;
    const int pix0   = pixb + m;
    const int pix1   = pixb + 16 + m;

    const float gx0 = grid_x[pix0], gy0 = grid_y[pix0];
    const float gx1 = grid_x[pix1], gy1 = grid_y[pix1];

    // A-coefficient pair offset (rows >= 3 read the zeroed pair at slot 6)
    const int aOff = (m < 3) ? 2 * m : 6;

    v8f acc0 = {};
    v8f acc1 = {};
    #pragma unroll 2
    for (int i = 0; i < NG / 2; ++i) {
        const int g = 2 * i + hi;          // this lane's gabor for this K-chunk
        const float4 p0 = *(const float4*)&sp[g * 8];
        const float4 p1 = *(const float4*)&sp[g * 8 + 4];
        const v2f    a  = *(const v2f*)&sc[g * 8 + aOff];   // one ds_load_b64

        // pixel set 0
        float dx = gx0 - p0.x, dy = gy0 - p0.y;
        float xr = dx * p0.z + dy * p0.w;
        float yr = dy * p0.z - dx * p0.w;
        float e  = fminf(fmaxf(-(xr * xr) * p1.x - (yr * yr) * p1.y,
                               -CLIP80L2), CLIP80L2);
        float gs = __builtin_amdgcn_exp2f(e);        // v_exp_f32
        float t  = p1.z * xr;                        // (freq/2pi) * x_rot
        v2f b0;
        b0.x = gs * __builtin_amdgcn_cosf(t);        // v_cos_f32: cos(2pi*t)
        b0.y = gs * __builtin_amdgcn_sinf(t);        // v_sin_f32

        // pixel set 1 (same gabor params, independent TRANS chain)
        dx = gx1 - p0.x; dy = gy1 - p0.y;
        xr = dx * p0.z + dy * p0.w;
        yr = dy * p0.z - dx * p0.w;
        e  = fminf(fmaxf(-(xr * xr) * p1.x - (yr * yr) * p1.y,
                         -CLIP80L2), CLIP80L2);
        gs = __builtin_amdgcn_exp2f(e);
        t  = p1.z * xr;
        v2f b1;
        b1.x = gs * __builtin_amdgcn_cosf(t);
        b1.y = gs * __builtin_amdgcn_sinf(t);

        // D = A(16x4) x B(4x16) + C, full f32
        acc0 = __builtin_amdgcn_wmma_f32_16x16x4_f32(
            false, a, false, b0, (short)0, acc0, false, false);
        acc1 = __builtin_amdgcn_wmma_f32_16x16x4_f32(
            false, a, false, b1, (short)0, acc1, false, false);
    }

    // Channels 0..2 live in acc[0..2] of lanes 0..15 (C/D layout: VGPR r -> M=r, N=lane).
    if (hi == 0) {
        out[0 * HW + pix0] = fminf(fmaxf(acc0[0], -1.f), 1.f);
        out[1 * HW + pix0] = fminf(fmaxf(acc0[1], -1.f), 1.f);
        out[2 * HW + pix0] = fminf(fmaxf(acc0[2], -1.f), 1.f);
        out[0 * HW + pix1] = fminf(fmaxf(acc1[0], -1.f), 1.f);
        out[1 * HW + pix1] = fminf(fmaxf(acc1[1], -1.f), 1.f);
        out[2 * HW + pix1] = fminf(fmaxf(acc1[2], -1.f), 1.f);
    }
}

extern "C" void kernel_launch(void* const* d_in, const int* in_sizes, int n_in,
                              void* d_out, int out_size, void* d_ws, size_t ws_size,
                              hipStream_t stream) {
    (void)in_sizes; (void)n_in; (void)d_ws; (void)ws_size; (void)out_size;
    const float* grid_x    = (const float*)d_in[0];
    const float* grid_y    = (const float*)d_in[1];
    const float* u         = (const float*)d_in[2];
    const float* v         = (const float*)d_in[3];
    const float* theta     = (const float*)d_in[4];
    const float* rel_sigma = (const float*)d_in[5];
    const float* rel_freq  = (const float*)d_in[6];
    const float* gamma_    = (const float*)d_in[7];
    const float* psi       = (const float*)d_in[8];
    const float* amplitude = (const float*)d_in[9];
    float* out             = (float*)d_out;

    // 262144 pixels / (8 waves * 32 pixels) = 1024 blocks; exact cover -> EXEC all-1s.
    gabor_wmma_kernel<<<HW / 256, 256, 0, stream>>>(
        grid_x, grid_y, u, v, theta, rel_sigma, rel_freq, gamma_, psi, amplitude, out);
}